// UniformShardedEmbeddingBags_23751169147034
// MI455X (gfx1250) — compile-verified
//
#include <hip/hip_runtime.h>
#include <stdint.h>

// Problem constants from the reference: T=32 tables, D=64 features.
#define TBL            32
#define DDIM           64
#define ROW_SHIFT      13            // T*D*4 = 8192 = 1<<13 bytes between idx rows
#define BAGS_PER_BLOCK 16
#define NTHREADS       256           // 8 wave32 per block, 16 lanes per bag
#define MAX_STAGE      64            // staged indices per bag in LDS (L=50 here)

struct alignas(16) F4 { float x, y, z, w; };

__device__ __forceinline__ void add4(F4& a, const F4 b) {
    a.x += b.x; a.y += b.y; a.z += b.z; a.w += b.w;
}

__global__ __launch_bounds__(NTHREADS)
void embag_pool_kernel(const float* __restrict__ weights,
                       const int*   __restrict__ feats,
                       const int*   __restrict__ offs,
                       float*       __restrict__ out,
                       int num_bags)
{
    __shared__ int idxbuf[BAGS_PER_BLOCK][MAX_STAGE];

    const int tid      = threadIdx.x;
    const int bagLocal = tid >> 4;        // 0..15
    const int lane16   = tid & 15;        // 0..15, owns a float4 slice of D
    const int bag      = blockIdx.x * BAGS_PER_BLOCK + bagLocal;
    if (bag >= num_bags) return;

    const int start = offs[bag];
    const int end   = offs[bag + 1];
    const int cnt   = end - start;
    const int cntS  = (cnt > MAX_STAGE) ? MAX_STAGE : cnt;

    // --- CDNA5 async global->LDS: stage this bag's indices once (ASYNCcnt path).
    for (int r = lane16; r < cntS; r += 16) {
        unsigned ldsoff = (unsigned)(uintptr_t)&idxbuf[bagLocal][r];
        uint64_t gaddr  = (uint64_t)(uintptr_t)(feats + start + r);
        asm volatile("global_load_async_to_lds_b32 %0, %1, off"
                     :: "v"(ldsoff), "v"(gaddr) : "memory");
    }
    asm volatile("s_wait_asynccnt 0" ::: "memory");

    // Per-lane byte offset inside a row: table = bag % 32, lane16 picks the
    // float4.  All row gathers become  s[weights] + u32_voffset  (fits 32 bits:
    // max ~819MB), i.e. SADDR-form global_load_b128.
    const unsigned laneoff =
        (unsigned)(((bag & (TBL - 1)) * DDIM + lane16 * 4) * 4);
    const char* wb = (const char*)weights;

    F4 acc = {0.f, 0.f, 0.f, 0.f};
    int j = 0;

    // Hot loop over staged indices: two ds_load_b128 broadcasts fetch 8
    // indices, then 8 independent global_load_b128 gathers stay in flight
    // (each row = 16 lanes x 16B = 256B = 2 full cachelines, coalesced).
    const int4* ivp = (const int4*)&idxbuf[bagLocal][0];
    for (; j + 8 <= cntS; j += 8) {
        const int4 ia = ivp[(j >> 2) + 0];
        const int4 ib = ivp[(j >> 2) + 1];
        F4 r0 = *(const F4*)(wb + (size_t)(laneoff + ((unsigned)ia.x << ROW_SHIFT)));
        F4 r1 = *(const F4*)(wb + (size_t)(laneoff + ((unsigned)ia.y << ROW_SHIFT)));
        F4 r2 = *(const F4*)(wb + (size_t)(laneoff + ((unsigned)ia.z << ROW_SHIFT)));
        F4 r3 = *(const F4*)(wb + (size_t)(laneoff + ((unsigned)ia.w << ROW_SHIFT)));
        F4 r4 = *(const F4*)(wb + (size_t)(laneoff + ((unsigned)ib.x << ROW_SHIFT)));
        F4 r5 = *(const F4*)(wb + (size_t)(laneoff + ((unsigned)ib.y << ROW_SHIFT)));
        F4 r6 = *(const F4*)(wb + (size_t)(laneoff + ((unsigned)ib.z << ROW_SHIFT)));
        F4 r7 = *(const F4*)(wb + (size_t)(laneoff + ((unsigned)ib.w << ROW_SHIFT)));
        add4(r0, r1); add4(r2, r3); add4(r4, r5); add4(r6, r7);
        add4(r0, r2); add4(r4, r6);
        add4(r0, r4);
        add4(acc, r0);
    }
    // Staged tail (j in [48,50) here).
    for (; j < cntS; ++j) {
        const int i0 = idxbuf[bagLocal][j];
        F4 r = *(const F4*)(wb + (size_t)(laneoff + ((unsigned)i0 << ROW_SHIFT)));
        add4(acc, r);
    }
    // Unstaged overflow (cnt > MAX_STAGE): straight from global. Never taken
    // for this reference (L=50), kept for generality.
    for (int k = cntS; k < cnt; ++k) {
        const int i0 = feats[start + k];
        F4 r = *(const F4*)(wb + (size_t)(laneoff + ((unsigned)i0 << ROW_SHIFT)));
        add4(acc, r);
    }

    // One coalesced b128 store per lane: out[bag, :] is contiguous.
    *(F4*)(out + (size_t)bag * DDIM + lane16 * 4) = acc;
}

extern "C" void kernel_launch(void* const* d_in, const int* in_sizes, int n_in,
                              void* d_out, int out_size, void* d_ws, size_t ws_size,
                              hipStream_t stream) {
    const float* weights = (const float*)d_in[0];   // (E, T, D) f32
    const int*   feats   = (const int*)d_in[1];     // (B*T*L,) i32
    const int*   offs    = (const int*)d_in[2];     // (B*T+1,) i32
    float*       out     = (float*)d_out;           // (B, T, D) f32

    const int num_bags = in_sizes[2] - 1;           // B*T = 32768
    const int grid     = (num_bags + BAGS_PER_BLOCK - 1) / BAGS_PER_BLOCK;

    hipLaunchKernelGGL(embag_pool_kernel, dim3(grid), dim3(NTHREADS), 0, stream,
                       weights, feats, offs, out, num_bags);
}